// Self_Attention_Channel_57286273794063
// MI455X (gfx1250) — compile-verified
//
#include <hip/hip_runtime.h>
#include <hip/hip_bf16.h>

// Problem constants (from reference): B=8, C=512, CR=32, H*W=9216
#define BB      8
#define CC      512
#define CRR     32
#define NPLANE  9216           // H*W
#define NP4     2304           // NPLANE/4
#define NPLANES 4096           // B*C

// ws layout (floats):
//  [0)      sum_t   (4096)
//  [4096)   max_t
//  [8192)   sum_b
//  [12288)  max_b
//  [16384)  sum_m
//  [20480)  max_m
//  [24576)  f  (B*C)
//  [28672)  g
//  [32768)  h
//  [36864)  vvec (B*C)
// total = 40960 floats = 160 KB

typedef __attribute__((ext_vector_type(2))) float v2f;
typedef __attribute__((ext_vector_type(8))) float v8f;

// ---------------------------------------------------------------------------
// Kernel 1: per-(b,c) plane: ori = t+b+m (written to out), sum & max of each
// input over H*W.  One block per plane (36 KB per tensor), b128 streaming.
// ---------------------------------------------------------------------------
__global__ __launch_bounds__(256) void k_reduce_ori(
    const float* __restrict__ t, const float* __restrict__ b,
    const float* __restrict__ m, float* __restrict__ ori,
    float* __restrict__ ws) {
  const int p = blockIdx.x;                  // plane = bIdx*C + c
  const size_t base4 = (size_t)p * NP4;      // in float4 units
  const int tid = threadIdx.x;

  const float4* t4 = ((const float4*)t) + base4;
  const float4* b4 = ((const float4*)b) + base4;
  const float4* m4 = ((const float4*)m) + base4;
  float4* o4 = ((float4*)ori) + base4;

  float s0 = 0.f, s1 = 0.f, s2 = 0.f;
  float x0 = -INFINITY, x1 = -INFINITY, x2 = -INFINITY;

#pragma unroll
  for (int j = 0; j < 9; ++j) {
    const int idx = j * 256 + tid;           // 0..2303
    float4 a = t4[idx];
    float4 c = b4[idx];
    float4 d = m4[idx];
    s0 += a.x + a.y + a.z + a.w;
    s1 += c.x + c.y + c.z + c.w;
    s2 += d.x + d.y + d.z + d.w;
    x0 = fmaxf(x0, fmaxf(fmaxf(a.x, a.y), fmaxf(a.z, a.w)));
    x1 = fmaxf(x1, fmaxf(fmaxf(c.x, c.y), fmaxf(c.z, c.w)));
    x2 = fmaxf(x2, fmaxf(fmaxf(d.x, d.y), fmaxf(d.z, d.w)));
    float4 o;
    o.x = a.x + c.x + d.x;
    o.y = a.y + c.y + d.y;
    o.z = a.z + c.z + d.z;
    o.w = a.w + c.w + d.w;
    o4[idx] = o;
  }

  // wave32 shuffle reduction
#pragma unroll
  for (int off = 16; off > 0; off >>= 1) {
    s0 += __shfl_xor(s0, off, 32);
    s1 += __shfl_xor(s1, off, 32);
    s2 += __shfl_xor(s2, off, 32);
    x0 = fmaxf(x0, __shfl_xor(x0, off, 32));
    x1 = fmaxf(x1, __shfl_xor(x1, off, 32));
    x2 = fmaxf(x2, __shfl_xor(x2, off, 32));
  }

  __shared__ float rbuf[6][8];
  const int wave = tid >> 5;
  if ((tid & 31) == 0) {
    rbuf[0][wave] = s0; rbuf[1][wave] = s1; rbuf[2][wave] = s2;
    rbuf[3][wave] = x0; rbuf[4][wave] = x1; rbuf[5][wave] = x2;
  }
  __syncthreads();
  if (tid == 0) {
    float S0 = rbuf[0][0], S1 = rbuf[1][0], S2 = rbuf[2][0];
    float M0 = rbuf[3][0], M1 = rbuf[4][0], M2 = rbuf[5][0];
#pragma unroll
    for (int w = 1; w < 8; ++w) {
      S0 += rbuf[0][w]; S1 += rbuf[1][w]; S2 += rbuf[2][w];
      M0 = fmaxf(M0, rbuf[3][w]);
      M1 = fmaxf(M1, rbuf[4][w]);
      M2 = fmaxf(M2, rbuf[5][w]);
    }
    ws[p]               = S0;  // sum_t
    ws[4096 + p]        = M0;  // max_t
    ws[8192 + p]        = S1;  // sum_b
    ws[12288 + p]       = M1;  // max_b
    ws[16384 + p]       = S2;  // sum_m
    ws[20480 + p]       = M2;  // max_m
  }
}

// ---------------------------------------------------------------------------
// Kernel 2: channel-attention MLPs via V_WMMA_F32_16X16X4_F32.
// One wave per block; blocks 0/1/2 compute f/g/h for tensors t/b/m.
// A (16x512): rows 0..7 = avg pooled, rows 8..15 = max pooled (both pools
// share one accumulator chain; layer-2 linearity folds the two relu paths).
// ---------------------------------------------------------------------------
__global__ __launch_bounds__(32) void k_mlp_wmma(
    const float* __restrict__ kW1, const float* __restrict__ kW2,
    const float* __restrict__ qW1, const float* __restrict__ qW2,
    const float* __restrict__ vW1, const float* __restrict__ vW2,
    float* __restrict__ ws) {
  const int blk = blockIdx.x;  // 0:t->f (kW), 1:b->g (qW), 2:m->h (vW)
  const float* W1 = (blk == 0) ? kW1 : ((blk == 1) ? qW1 : vW1); // (CR=32, C=512)
  const float* W2 = (blk == 0) ? kW2 : ((blk == 1) ? qW2 : vW2); // (C=512, CR=32)
  const float* sumv = ws + (size_t)blk * 8192;
  const float* maxv = sumv + 4096;
  float* outv = ws + 24576 + (size_t)blk * 4096;

  const int lane = threadIdx.x;
  const int half = lane >> 4;        // 0: K+{0,1}, 1: K+{2,3}
  const int l16  = lane & 15;
  const int koff = half * 2;

  // A row for this lane: m = l16 (0..7 avg batches, 8..15 max batches)
  const float* Arow = (l16 < 8) ? (sumv + l16 * CC) : (maxv + (l16 - 8) * CC);
  const float ascale = (l16 < 8) ? (1.0f / (float)NPLANE) : 1.0f;
  const float* B0row = W1 + l16 * CC;          // n = l16        (tile 0)
  const float* B1row = W1 + (16 + l16) * CC;   // n = 16 + l16   (tile 1)

  v8f acc0 = {0.f, 0.f, 0.f, 0.f, 0.f, 0.f, 0.f, 0.f};
  v8f acc1 = {0.f, 0.f, 0.f, 0.f, 0.f, 0.f, 0.f, 0.f};

#pragma unroll 4
  for (int k = 0; k < CC; k += 4) {
    const int o = k + koff;
    v2f a;
    a.x = Arow[o] * ascale;
    a.y = Arow[o + 1] * ascale;
    v2f b0 = *(const v2f*)(B0row + o);
    v2f b1 = *(const v2f*)(B1row + o);
    acc0 = __builtin_amdgcn_wmma_f32_16x16x4_f32(
        false, a, false, b0, (short)0, acc0, false, false);
    acc1 = __builtin_amdgcn_wmma_f32_16x16x4_f32(
        false, a, false, b1, (short)0, acc1, false, false);
  }

  // relu and stage layer-1 result: hid[16][32]; rows 0..7 avg, 8..15 max
  __shared__ float hid[16 * 32];
#pragma unroll
  for (int vv = 0; vv < 8; ++vv) {
    const int row = vv + 8 * half;   // M = vv + 8*half
    hid[row * 32 + l16]      = fmaxf(acc0[vv], 0.f);
    hid[row * 32 + 16 + l16] = fmaxf(acc1[vv], 0.f);
  }
  __syncthreads();

  // Layer 2: out[b][n] = sum_j (hid[b][j]+hid[b+8][j]) * W2[n][j]
  const int arow = (l16 & 7) * 32;                 // clamp pad rows in-range
  const float amask = (l16 < 8) ? 1.0f : 0.0f;     // arithmetic pad (no EXEC div.)

  for (int nt = 0; nt < 32; ++nt) {
    const float* B2row = W2 + (size_t)(nt * 16 + l16) * CRR;  // row stride 32
    v8f acc = {0.f, 0.f, 0.f, 0.f, 0.f, 0.f, 0.f, 0.f};
#pragma unroll
    for (int kb = 0; kb < CRR; kb += 4) {
      const int j = kb + koff;
      v2f a2;
      a2.x = amask * (hid[arow + j]     + hid[arow + 256 + j]);
      a2.y = amask * (hid[arow + j + 1] + hid[arow + 256 + j + 1]);
      v2f b2 = *(const v2f*)(B2row + j);
      acc = __builtin_amdgcn_wmma_f32_16x16x4_f32(
          false, a2, false, b2, (short)0, acc, false, false);
    }
    if (half == 0) {  // lanes 0..15 hold M = vv (batches 0..7)
#pragma unroll
      for (int vv = 0; vv < 8; ++vv) {
        outv[vv * CC + nt * 16 + l16] = acc[vv];
      }
    }
  }
}

// ---------------------------------------------------------------------------
// Kernel 3: v[b,d] = sum_c h[b,c]*softmax_c(f[b,c]*g[b,d]).
// Per-column max of s is g_d*maxf or g_d*minf depending on sign(g_d).
// 16 blocks x 256 threads = 4096 (b,d) pairs; f/h staged in LDS.
// ---------------------------------------------------------------------------
__global__ __launch_bounds__(256) void k_attn(
    const float* __restrict__ ws, float* __restrict__ vout) {
  const float* f = ws + 24576;
  const float* g = ws + 28672;
  const float* h = ws + 32768;

  const int gid = blockIdx.x * 256 + threadIdx.x;  // 0..4095
  const int bI = gid >> 9;         // same for all threads in a block
  const int d  = gid & 511;
  const int tid = threadIdx.x;

  __shared__ float fsh[512];
  __shared__ float hsh[512];
  __shared__ float rmax[8], rmin[8];
  __shared__ float fm[2];

  const float* fb = f + bI * CC;
  const float* hb = h + bI * CC;
  const float f0 = fb[tid], f1 = fb[tid + 256];
  fsh[tid] = f0;       fsh[tid + 256] = f1;
  hsh[tid] = hb[tid];  hsh[tid + 256] = hb[tid + 256];

  float lmax = fmaxf(f0, f1);
  float lmin = fminf(f0, f1);
#pragma unroll
  for (int off = 16; off > 0; off >>= 1) {
    lmax = fmaxf(lmax, __shfl_xor(lmax, off, 32));
    lmin = fminf(lmin, __shfl_xor(lmin, off, 32));
  }
  if ((tid & 31) == 0) { rmax[tid >> 5] = lmax; rmin[tid >> 5] = lmin; }
  __syncthreads();
  if (tid == 0) {
    float a = rmax[0], i = rmin[0];
#pragma unroll
    for (int w = 1; w < 8; ++w) { a = fmaxf(a, rmax[w]); i = fminf(i, rmin[w]); }
    fm[0] = a; fm[1] = i;
  }
  __syncthreads();

  const float gd = g[bI * CC + d];
  const float mstar = fmaxf(gd * fm[0], gd * fm[1]);
  float num = 0.f, den = 0.f;
#pragma unroll 8
  for (int c = 0; c < CC; ++c) {
    const float e = __expf(fsh[c] * gd - mstar);
    den += e;
    num = fmaf(hsh[c], e, num);
  }
  vout[gid] = num / den;
}

// ---------------------------------------------------------------------------
// Kernel 4: y = ori + gamma * v[b,c]   (in-place on d_out; ori likely L2-hot)
// ---------------------------------------------------------------------------
__global__ __launch_bounds__(256) void k_bcast(
    float* __restrict__ out, const float* __restrict__ ws,
    const float* __restrict__ gamma) {
  const int p = blockIdx.x;
  const float add = gamma[0] * ws[36864 + p];
  float4* o4 = ((float4*)out) + (size_t)p * NP4;
  const int tid = threadIdx.x;
#pragma unroll
  for (int j = 0; j < 9; ++j) {
    const int idx = j * 256 + tid;
    float4 x = o4[idx];
    x.x += add; x.y += add; x.z += add; x.w += add;
    o4[idx] = x;
  }
}

extern "C" void kernel_launch(void* const* d_in, const int* in_sizes, int n_in,
                              void* d_out, int out_size, void* d_ws, size_t ws_size,
                              hipStream_t stream) {
  const float* t   = (const float*)d_in[0];
  const float* b   = (const float*)d_in[1];
  const float* m   = (const float*)d_in[2];
  const float* kW1 = (const float*)d_in[3];
  const float* kW2 = (const float*)d_in[4];
  const float* qW1 = (const float*)d_in[5];
  const float* qW2 = (const float*)d_in[6];
  const float* vW1 = (const float*)d_in[7];
  const float* vW2 = (const float*)d_in[8];
  const float* gamma = (const float*)d_in[9];

  float* out = (float*)d_out;
  float* ws  = (float*)d_ws;   // needs 160 KB

  k_reduce_ori<<<NPLANES, 256, 0, stream>>>(t, b, m, out, ws);
  k_mlp_wmma<<<3, 32, 0, stream>>>(kW1, kW2, qW1, qW2, vW1, vW2, ws);
  k_attn<<<16, 256, 0, stream>>>(ws, ws + 36864);
  k_bcast<<<NPLANES, 256, 0, stream>>>(out, ws, gamma);
}